// get_knearest_neighbors_torch_xy_43516608643712
// MI455X (gfx1250) — compile-verified
//
#include <hip/hip_runtime.h>
#include <hip/hip_bf16.h>

// KNN-16 deltas for 16384 2-D points.
// Strategy: key(i,j) = ||p_j||^2 - 2*(p_i . p_j)  (same ordering as d2 per row)
// computed as a rank-3 product folded into V_WMMA_F32_16X16X4_F32:
//   A[i] = [-2x_i, -2y_i, 1, 0]   (16x4)
//   B[j] = [ x_j,   y_j, sq_j, 0] (4x16)
// One WMMA per 16x16 tile yields the full comparison-key tile.
// Per-wave: tile -> LDS -> row-major re-own -> register-resident sorted top-16.

#define NPTS 16384
#define KNN  16
#define NT   (NPTS / 16)   // 1024 column tiles
#define WPB  8             // waves per block (wave32)

typedef __attribute__((ext_vector_type(2))) float v2f;
typedef __attribute__((ext_vector_type(8))) float v8f;

__global__ __launch_bounds__(WPB * 32)
void knn16_wmma_kernel(const float* __restrict__ p, float* __restrict__ out) {
  __shared__ float s_tile[WPB][16][16];     // per-wave 16x16 key tile
  __shared__ float s_key[WPB][32][KNN];     // per-lane sorted partial lists
  __shared__ int   s_idx[WPB][32][KNN];

  const int lane  = threadIdx.x & 31;
  const int wv    = threadIdx.x >> 5;
  const int itile = blockIdx.x * WPB + wv;  // 0..1023
  const int i0    = itile * 16;

  const int r    = lane & 15;               // row this lane owns for selection
  const int cb   = (lane >> 4) * 8;         // column sub-range base (0 or 8)
  const int irow = i0 + r;                  // global query index for this lane
  const bool hi  = lane >= 16;

  // ---- A operand (f32 16x4 layout: v0 = K0|K2, v1 = K1|K3 across lane halves)
  const float xi = p[3 * irow + 0];
  const float yi = p[3 * irow + 1];
  v2f A;
  A.x = hi ? 1.0f : (-2.0f * xi);   // K0 = -2x   | K2 = 1
  A.y = hi ? 0.0f : (-2.0f * yi);   // K1 = -2y   | K3 = 0

  // ---- private top-K list, sorted ascending; keys[KNN-1] is the current worst
  float keys[KNN];
  int   idxs[KNN];
#pragma unroll
  for (int t = 0; t < KNN; ++t) { keys[t] = 3.0e38f; idxs[t] = 0; }

  for (int jt = 0; jt < NT; ++jt) {
    const int j0 = jt * 16;
    const int jl = j0 + r;
    const float xj = p[3 * jl + 0];
    const float yj = p[3 * jl + 1];

    // prefetch the next column tile (global_prefetch_b8; L2-resident anyway)
    __builtin_prefetch(&p[3 * ((j0 + 16) & (NPTS - 1))], 0, 0);

    // ---- B operand (4x16: v0 = K0|K2, v1 = K1|K3 across lane halves)
    v2f B;
    B.x = hi ? (xj * xj + yj * yj) : xj;   // K0 = x  | K2 = sq
    B.y = hi ? 0.0f : yj;                  // K1 = y  | K3 = 0

    v8f C = {};
    C = __builtin_amdgcn_wmma_f32_16x16x4_f32(false, A, false, B,
                                              (short)0, C, false, false);

    // C[s] holds key(row = cb + s, col = r) -> scatter to LDS row-major
#pragma unroll
    for (int s = 0; s < 8; ++s)
      s_tile[wv][cb + s][r] = C[s];

    // same-wave cross-lane LDS RAW: drain DS ops
    asm volatile("s_wait_dscnt 0" ::: "memory");

    // re-own row-major: lane handles row r, cols cb..cb+7 of this tile
#pragma unroll
    for (int c = 0; c < 8; ++c) {
      const float key = s_tile[wv][r][cb + c];
      const int   j   = j0 + cb + c;
      if (j != irow && key < keys[KNN - 1]) {      // fast path: 1 compare
        float ck = key; int ci = j;
#pragma unroll
        for (int t = 0; t < KNN; ++t) {            // unrolled sorted insert
          const bool  lt = ck < keys[t];
          const float tv = keys[t];
          const int   ti = idxs[t];
          keys[t] = lt ? ck : tv;
          idxs[t] = lt ? ci : ti;
          ck = lt ? tv : ck;
          ci = lt ? ti : ci;
        }
      }
    }
  }

  // ---- publish per-lane lists, then merge lane L with lane L+16 per row
#pragma unroll
  for (int t = 0; t < KNN; ++t) {
    s_key[wv][lane][t] = keys[t];
    s_idx[wv][lane][t] = idxs[t];
  }
  asm volatile("s_wait_dscnt 0" ::: "memory");

  if (lane < 16) {
    int pa = 0, pb = 0;
    for (int t = 0; t < KNN; ++t) {
      const float ka = s_key[wv][lane][pa];
      const float kb = s_key[wv][lane + 16][pb];
      const int   ja = s_idx[wv][lane][pa];
      const int   jb = s_idx[wv][lane + 16][pb];
      const bool takeA = (ka < kb) || (ka == kb && ja <= jb);  // tie: low index
      const int j = takeA ? ja : jb;
      pa += takeA ? 1 : 0;
      pb += takeA ? 0 : 1;
      const float dx = xi - p[3 * j + 0];
      const float dy = yi - p[3 * j + 1];
      out[(irow * KNN + t) * 2 + 0] = dx;
      out[(irow * KNN + t) * 2 + 1] = dy;
    }
  }
}

extern "C" void kernel_launch(void* const* d_in, const int* in_sizes, int n_in,
                              void* d_out, int out_size, void* d_ws, size_t ws_size,
                              hipStream_t stream) {
  const float* p = (const float*)d_in[0];   // (16384, 3) float32
  float* out = (float*)d_out;               // (16384, 16, 2) float32
  (void)in_sizes; (void)n_in; (void)out_size; (void)d_ws; (void)ws_size;

  dim3 grid(NPTS / 16 / WPB);   // 128 blocks
  dim3 block(WPB * 32);         // 256 threads = 8 wave32
  hipLaunchKernelGGL(knn16_wmma_kernel, grid, block, 0, stream, p, out);
}